// Decoder_2164663517582
// MI455X (gfx1250) — compile-verified
//
#include <hip/hip_runtime.h>
#include <math.h>

#define HID   512
#define EMB   256
#define LEN   32
#define BATCH 32
#define VOCAB 32000

typedef float v2f __attribute__((ext_vector_type(2)));
typedef float v8f __attribute__((ext_vector_type(8)));

// D = A(16x4) * B(4x16) + C, full fp32 (V_WMMA_F32_16X16X4_F32)
static __device__ __forceinline__ v8f wmma4(v2f a, v2f b, v8f c) {
  return __builtin_amdgcn_wmma_f32_16x16x4_f32(
      /*neg_a=*/false, a, /*neg_b=*/false, b,
      /*c_mod=*/(short)0, c, /*reuse_a=*/false, /*reuse_b=*/false);
}

static __device__ __forceinline__ float sigm(float x) {
  return 1.0f / (1.0f + expf(-x));
}

// ---------------- init: zero cell state ----------------
__global__ void k_zero(float* __restrict__ p) {
  p[blockIdx.x * 256 + threadIdx.x] = 0.0f;
}

// ---------------- key_proj = enc @ Ua^T + Ua_b  ([1024,512]x[512,512]) ----------------
__global__ void k_keyproj(const float* __restrict__ enc, const float* __restrict__ Ua_w,
                          const float* __restrict__ Ua_b, float* __restrict__ keyp) {
  int tid = threadIdx.x, wave = tid >> 5, lane = tid & 31;
  int ln = lane & 15, koff = (lane >> 4) << 1;
  int tile = blockIdx.x * 8 + wave;        // 2048 tiles total
  int mt = tile & 63, nt = tile >> 6;      // 64 M-tiles x 32 N-tiles
  int m0 = mt * 16, n0 = nt * 16;
  const float* xrow = enc + (m0 + ln) * HID;
  const float* wrow = Ua_w + (n0 + ln) * HID;
  v8f acc = {};
  for (int k = 0; k < HID; k += 4) {
    v2f a = *(const v2f*)(xrow + k + koff);
    v2f b = *(const v2f*)(wrow + k + koff);
    acc = wmma4(a, b, acc);
  }
  float bias = Ua_b[n0 + ln];
  int mo = m0 + ((lane >> 4) << 3);
#pragma unroll
  for (int r = 0; r < 8; ++r)
    keyp[(mo + r) * HID + n0 + ln] = acc[r] + bias;
}

// ---------------- gates GEMM + LSTM cell (one WG per 16 hidden units) ----------------
__global__ void k_gates(const float* __restrict__ emb, const int* __restrict__ tgt,
                        const int* __restrict__ sosp, int t,
                        const float* __restrict__ hprev,
                        const float* __restrict__ W_ih, const float* __restrict__ W_hh,
                        const float* __restrict__ b_ih, const float* __restrict__ b_hh,
                        float* __restrict__ cbuf, float* __restrict__ hout) {
  __shared__ float gl[4][32][16];          // [gate][row][unit]
  int tid = threadIdx.x, wave = tid >> 5, lane = tid & 31;
  int ln = lane & 15, koff = (lane >> 4) << 1;
  int j0 = blockIdx.x * 16;                // hidden-unit slab
  int mt = wave & 1, gate = wave >> 1;     // 2 M-tiles x 4 gates = 8 waves
  int m0 = mt * 16;
  int m = m0 + ln;
  int idx = (t == 0) ? sosp[0] : tgt[m * LEN + (t - 1)];
  const float* xrow = emb + (long)idx * EMB;
  int n = gate * HID + j0 + ln;            // row of W_ih / W_hh
  const float* w1 = W_ih + (long)n * EMB;
  const float* w2 = W_hh + (long)n * HID;
  const float* hrow = hprev + m * HID;
  v8f acc = {};
  for (int k = 0; k < EMB; k += 4) {
    v2f a = *(const v2f*)(xrow + k + koff);
    v2f b = *(const v2f*)(w1 + k + koff);
    acc = wmma4(a, b, acc);
  }
  for (int k = 0; k < HID; k += 4) {
    v2f a = *(const v2f*)(hrow + k + koff);
    v2f b = *(const v2f*)(w2 + k + koff);
    acc = wmma4(a, b, acc);
  }
  float bias = b_ih[n] + b_hh[n];
  int mo = m0 + ((lane >> 4) << 3);
#pragma unroll
  for (int r = 0; r < 8; ++r)
    gl[gate][mo + r][ln] = acc[r] + bias;
  __syncthreads();
  for (int e = tid; e < 32 * 16; e += 256) {
    int row = e >> 4, col = e & 15;
    int j = j0 + col;
    float iv = sigm(gl[0][row][col]);
    float fv = sigm(gl[1][row][col]);
    float gv = tanhf(gl[2][row][col]);
    float ov = sigm(gl[3][row][col]);
    float cn = fv * cbuf[row * HID + j] + iv * gv;
    cbuf[row * HID + j] = cn;
    hout[row * HID + j] = ov * tanhf(cn);
  }
}

// ---------------- q = h @ Wa^T + Wa_b ----------------
__global__ void k_q(const float* __restrict__ h, const float* __restrict__ Wa_w,
                    const float* __restrict__ Wa_b, float* __restrict__ q) {
  int tid = threadIdx.x, wave = tid >> 5, lane = tid & 31;
  int ln = lane & 15, koff = (lane >> 4) << 1;
  int tile = blockIdx.x * 8 + wave;        // 64 tiles: 2 M x 32 N
  int mt = tile & 1, nt = tile >> 1;
  int m0 = mt * 16, n0 = nt * 16;
  const float* xrow = h + (m0 + ln) * HID;
  const float* wrow = Wa_w + (n0 + ln) * HID;
  v8f acc = {};
  for (int k = 0; k < HID; k += 4) {
    v2f a = *(const v2f*)(xrow + k + koff);
    v2f b = *(const v2f*)(wrow + k + koff);
    acc = wmma4(a, b, acc);
  }
  float bias = Wa_b[n0 + ln];
  int mo = m0 + ((lane >> 4) << 3);
#pragma unroll
  for (int r = 0; r < 8; ++r)
    q[(mo + r) * HID + n0 + ln] = acc[r] + bias;
}

// ---------------- additive attention + softmax + context (one WG per batch) ----------------
__global__ void k_attn(const float* __restrict__ q, const float* __restrict__ keyp,
                       const float* __restrict__ va, const float* __restrict__ va_b,
                       const int* __restrict__ mask, const float* __restrict__ enc,
                       float* __restrict__ ctx) {
  __shared__ float part[32][8];
  __shared__ float attn[32];
  int b = blockIdx.x, tid = threadIdx.x;
  int l = tid >> 3, sub = tid & 7;
  const float* qb = q + b * HID;
  const float* kp = keyp + (b * LEN + l) * HID;
  float p = 0.0f;
  for (int hh = sub; hh < HID; hh += 8)
    p += va[hh] * tanhf(qb[hh] + kp[hh]);
  part[l][sub] = p;
  __syncthreads();
  if (tid < 32) {                          // wave 0: per-lane score l=tid
    float s = va_b[0];
#pragma unroll
    for (int u = 0; u < 8; ++u) s += part[tid][u];
    if (mask[b * LEN + tid] == 0) s = -1e10f;
    float mx = s;
    for (int o = 16; o > 0; o >>= 1) mx = fmaxf(mx, __shfl_xor(mx, o, 32));
    float e = expf(s - mx);
    float sm = e;
    for (int o = 16; o > 0; o >>= 1) sm += __shfl_xor(sm, o, 32);
    attn[tid] = e / sm;
  }
  __syncthreads();
  for (int hh = tid; hh < HID; hh += 256) {
    float a = 0.0f;
    for (int l2 = 0; l2 < LEN; ++l2)
      a += attn[l2] * enc[(b * LEN + l2) * HID + hh];
    ctx[b * HID + hh] = a;
  }
}

// ---------------- act = tanh([ctx,h] @ Wc^T + Wc_b) ----------------
__global__ void k_wc(const float* __restrict__ ctx, const float* __restrict__ h,
                     const float* __restrict__ Wc_w, const float* __restrict__ Wc_b,
                     float* __restrict__ act) {
  int tid = threadIdx.x, wave = tid >> 5, lane = tid & 31;
  int ln = lane & 15, koff = (lane >> 4) << 1;
  int tile = blockIdx.x * 8 + wave;        // 64 tiles
  int mt = tile & 1, nt = tile >> 1;
  int m0 = mt * 16, n0 = nt * 16;
  const float* crow = ctx + (m0 + ln) * HID;
  const float* hrow = h + (m0 + ln) * HID;
  const float* wrow = Wc_w + (long)(n0 + ln) * (2 * HID);
  v8f acc = {};
  for (int k = 0; k < HID; k += 4) {       // first half of cat: ctx
    v2f a = *(const v2f*)(crow + k + koff);
    v2f b = *(const v2f*)(wrow + k + koff);
    acc = wmma4(a, b, acc);
  }
  for (int k = 0; k < HID; k += 4) {       // second half of cat: h
    v2f a = *(const v2f*)(hrow + k + koff);
    v2f b = *(const v2f*)(wrow + HID + k + koff);
    acc = wmma4(a, b, acc);
  }
  float bias = Wc_b[n0 + ln];
  int mo = m0 + ((lane >> 4) << 3);
#pragma unroll
  for (int r = 0; r < 8; ++r)
    act[(mo + r) * HID + n0 + ln] = tanhf(acc[r] + bias);
}

// ---------------- out[:,t,:] = act @ Wh^T + Wh_b  (N = 32000) ----------------
__global__ void k_vocab(const float* __restrict__ act, const float* __restrict__ Wh_w,
                        const float* __restrict__ Wh_b, float* __restrict__ out, int t) {
  int tid = threadIdx.x, wave = tid >> 5, lane = tid & 31;
  int ln = lane & 15, koff = (lane >> 4) << 1;
  int nbase = blockIdx.x * 64;             // 500 WGs * 64 cols = 32000
  int mt = wave & 1, nt = wave >> 1;       // 2 M-tiles x 4 N-tiles
  int m0 = mt * 16, n0 = nbase + nt * 16;
  const float* xrow = act + (m0 + ln) * HID;
  const float* wrow = Wh_w + (long)(n0 + ln) * HID;
  v8f acc = {};
  for (int k = 0; k < HID; k += 4) {
    v2f a = *(const v2f*)(xrow + k + koff);
    v2f b = *(const v2f*)(wrow + k + koff);
    acc = wmma4(a, b, acc);
  }
  float bias = Wh_b[n0 + ln];
  int mo = m0 + ((lane >> 4) << 3);
#pragma unroll
  for (int r = 0; r < 8; ++r)
    out[((long)(mo + r) * LEN + t) * VOCAB + n0 + ln] = acc[r] + bias;
}

extern "C" void kernel_launch(void* const* d_in, const int* in_sizes, int n_in,
                              void* d_out, int out_size, void* d_ws, size_t ws_size,
                              hipStream_t stream) {
  (void)in_sizes; (void)n_in; (void)out_size; (void)ws_size;
  const int*   mask     = (const int*)d_in[0];
  const float* enc      = (const float*)d_in[1];
  const float* enc_last = (const float*)d_in[2];   // [1,B,H] -> [B,H]
  const int*   tgt      = (const int*)d_in[3];
  const int*   sos      = (const int*)d_in[4];
  const float* emb      = (const float*)d_in[5];
  const float* W_ih     = (const float*)d_in[6];
  const float* W_hh     = (const float*)d_in[7];
  const float* b_ih     = (const float*)d_in[8];
  const float* b_hh     = (const float*)d_in[9];
  const float* Wa_w     = (const float*)d_in[10];
  const float* Wa_b     = (const float*)d_in[11];
  const float* Ua_w     = (const float*)d_in[12];
  const float* Ua_b     = (const float*)d_in[13];
  const float* va_w     = (const float*)d_in[14];
  const float* va_b     = (const float*)d_in[15];
  const float* Wc_w     = (const float*)d_in[16];
  const float* Wc_b     = (const float*)d_in[17];
  const float* Wh_w     = (const float*)d_in[18];
  const float* Wh_b     = (const float*)d_in[19];
  float* out = (float*)d_out;

  // workspace layout (floats)
  float* ws     = (float*)d_ws;
  float* keyp   = ws;                       // 1024*512
  float* hbuf   = keyp + 1024 * 512;        // 2 * 32*512 (double-buffered h)
  float* cbuf   = hbuf + 2 * BATCH * HID;   // 32*512
  float* qbuf   = cbuf + BATCH * HID;       // 32*512
  float* ctxbuf = qbuf + BATCH * HID;       // 32*512
  float* actbuf = ctxbuf + BATCH * HID;     // 32*512

  k_zero<<<64, 256, 0, stream>>>(cbuf);                       // c0 = 0
  k_keyproj<<<256, 256, 0, stream>>>(enc, Ua_w, Ua_b, keyp);  // Ua(h_enc) once

  for (int t = 0; t < LEN; ++t) {
    const float* hprev = (t == 0) ? enc_last : (hbuf + ((t + 1) & 1) * BATCH * HID);
    float* hout = hbuf + (t & 1) * BATCH * HID;
    k_gates<<<32, 256, 0, stream>>>(emb, tgt, sos, t, hprev, W_ih, W_hh, b_ih, b_hh,
                                    cbuf, hout);
    k_q<<<8, 256, 0, stream>>>(hout, Wa_w, Wa_b, qbuf);
    k_attn<<<32, 256, 0, stream>>>(qbuf, keyp, va_w, va_b, mask, enc, ctxbuf);
    k_wc<<<8, 256, 0, stream>>>(ctxbuf, hout, Wc_w, Wc_b, actbuf);
    k_vocab<<<500, 256, 0, stream>>>(actbuf, Wh_w, Wh_b, out, t);
  }
}